// VectorQuantizerEMA_35983236006235
// MI455X (gfx1250) — compile-verified
//
#include <hip/hip_runtime.h>
#include <math.h>

// Problem constants (from reference)
#define KC   512          // num codes
#define DD   64           // embedding dim
#define BB   32           // batch
#define LL   4096         // sequence length
#define NR   (BB * LL)    // 131072 flattened rows
#define DECAYF 0.99f
#define BETAF  0.25f
#define EPSF   1e-5f

// d_out layout: [z_q_st : 8388608][vq_loss : 1][perplexity : 1][indices-as-float : 131072]
#define OUT_ZQ   0
#define OUT_LOSS (BB * DD * LL)          // 8388608
#define OUT_PERP (OUT_LOSS + 1)
#define OUT_IDX  (OUT_LOSS + 2)

// workspace layout (float units)
#define WS_ENORM  0                      // 512
#define WS_COUNTS 512                    // 512
#define WS_ESUM   1024                   // 512*64 = 32768
#define WS_NEWEMB 33792                  // 32768
#define WS_LOSS   66560                  // 1
#define WS_IDX    66564                  // 131072 ints

typedef __attribute__((ext_vector_type(2))) float v2f;
typedef __attribute__((ext_vector_type(8))) float v8f;

// ---------------- zero scratch (counts, esum, loss) ----------------
__global__ __launch_bounds__(256) void vq_zero(float* __restrict__ ws) {
    int t = blockIdx.x * 256 + threadIdx.x;
    if (t < 33280) ws[WS_COUNTS + t] = 0.0f;   // counts + esum
    else if (t == 33280) ws[WS_LOSS] = 0.0f;
}

// ---------------- ||e_k||^2 ----------------
__global__ __launch_bounds__(256) void vq_enorm(const float* __restrict__ emb,
                                                float* __restrict__ e_norm) {
    int k = blockIdx.x * 256 + threadIdx.x;
    if (k < KC) {
        const float* row = emb + (size_t)k * DD;
        float s = 0.0f;
        #pragma unroll 8
        for (int d = 0; d < DD; ++d) { float v = row[d]; s += v * v; }
        e_norm[k] = s;
    }
}

// ---------------- WMMA distance + argmin ----------------
// One wave32 per 16-row tile; 8 waves per block.
// dist = ||z||^2 - 2 z.e + ||e||^2 ; argmin over 512 codes.
__global__ __launch_bounds__(256) void vq_argmin_wmma(
    const float* __restrict__ z,          // z_e [B, D, L]
    const float* __restrict__ emb,        // [K, D]
    const float* __restrict__ e_norm,     // [K]
    int*   __restrict__ idxout,           // [N]
    float* __restrict__ counts)           // [K]
{
    const int lane = threadIdx.x & 31;
    const int wave = threadIdx.x >> 5;
    const int tile = blockIdx.x * 8 + wave;      // 1024 blocks * 8 waves = 8192 tiles
    const int rowBase = tile * 16;
    const int b  = rowBase >> 12;                // rowBase / LL (tile never crosses batch)
    const int l0 = rowBase & (LL - 1);
    const int m   = lane & 15;
    const int khi = lane >> 4;                   // 0: K%4 in {0,1}; 1: K%4 in {2,3}

    // ---- load A fragments for 16 k-steps (D=64 = 16 * 4) ----
    // A row (M = rowBase+m) column d lives at z[(b*DD + d)*LL + l0 + m]
    v2f a[16];
    const float* zb = z + ((size_t)b * DD) * LL + (size_t)(l0 + m);
    #pragma unroll
    for (int kk = 0; kk < 16; ++kk) {
        const int k0 = kk * 4 + khi * 2;
        a[kk].x = zb[(size_t)k0 * LL];
        a[kk].y = zb[(size_t)(k0 + 1) * LL];
    }

    // ---- ||z_row||^2 : per-lane partial, combine halves, redistribute ----
    float ps = 0.0f;
    #pragma unroll
    for (int kk = 0; kk < 16; ++kk) ps += a[kk].x * a[kk].x + a[kk].y * a[kk].y;
    ps += __shfl_xor(ps, 16);                    // lane now holds znorm[row m]
    float zn8[8];
    #pragma unroll
    for (int r = 0; r < 8; ++r)
        zn8[r] = __shfl(ps, r + khi * 8);        // znorm for C-slot rows

    float minv[8];
    int   mini[8];
    #pragma unroll
    for (int r = 0; r < 8; ++r) { minv[r] = 3.4e38f; mini[r] = 0; }

    // ---- sweep 32 column tiles of 16 codes ----
    for (int nt = 0; nt < 32; ++nt) {
        const int colBase = nt * 16;
        const float* eb = emb + (size_t)(colBase + m) * DD + khi * 2;

        v8f c = {0.f, 0.f, 0.f, 0.f, 0.f, 0.f, 0.f, 0.f};
        #pragma unroll
        for (int kk = 0; kk < 16; ++kk) {
            v2f bf;
            bf.x = eb[kk * 4];
            bf.y = eb[kk * 4 + 1];
            c = __builtin_amdgcn_wmma_f32_16x16x4_f32(
                    false, a[kk], false, bf, (short)0, c, false, false);
        }

        const float en  = e_norm[colBase + m];
        const int   col = colBase + m;
        #pragma unroll
        for (int r = 0; r < 8; ++r) {
            float dist = zn8[r] - 2.0f * c[r] + en;
            if (dist < minv[r]) { minv[r] = dist; mini[r] = col; }
        }
    }

    // ---- cross-lane argmin within each 16-lane half (first-min tie-break) ----
    #pragma unroll
    for (int off = 1; off < 16; off <<= 1) {
        #pragma unroll
        for (int r = 0; r < 8; ++r) {
            float ov = __shfl_xor(minv[r], off);
            int   oi = __shfl_xor(mini[r], off);
            if (ov < minv[r] || (ov == minv[r] && oi < mini[r])) {
                minv[r] = ov; mini[r] = oi;
            }
        }
    }

    if (m == 0) {   // lanes 0 (rows 0..7) and 16 (rows 8..15)
        #pragma unroll
        for (int r = 0; r < 8; ++r) {
            const int row = rowBase + r + khi * 8;
            idxout[row] = mini[r];
            atomicAdd(&counts[mini[r]], 1.0f);
        }
    }
}

// ---------------- embed_sum scatter: esum[idx[n], d] += z[n, d] ----------------
__global__ __launch_bounds__(256) void vq_scatter(
    const float* __restrict__ z, const int* __restrict__ idx,
    float* __restrict__ esum)
{
    const int t = blockIdx.x * 256 + threadIdx.x;  // over N*64 elements
    const int n = t >> 6;
    const int d = t & 63;
    const int b = n >> 12;
    const int l = n & (LL - 1);
    const float v = z[((size_t)(b * DD + d)) * LL + l];
    atomicAdd(&esum[(size_t)idx[n] * DD + d], v);
}

// ---------------- EMA codebook update (single block, 512 threads) ----------------
__global__ __launch_bounds__(512) void vq_ema(
    const float* __restrict__ cluster_size, const float* __restrict__ emb_avg,
    const float* __restrict__ counts, const float* __restrict__ esum,
    float* __restrict__ newemb)
{
    __shared__ float red[512];
    const int k = threadIdx.x;
    const float ncs = cluster_size[k] * DECAYF + (1.0f - DECAYF) * counts[k];
    red[k] = ncs;
    __syncthreads();
    for (int s = 256; s > 0; s >>= 1) {
        if (k < s) red[k] += red[k + s];
        __syncthreads();
    }
    const float n  = red[0];
    const float cs = (ncs + EPSF) / (n + KC * EPSF) * n;
    const float inv = 1.0f / cs;
    #pragma unroll 8
    for (int d = 0; d < DD; ++d) {
        const int o = k * DD + d;
        newemb[o] = (emb_avg[o] * DECAYF + (1.0f - DECAYF) * esum[o]) * inv;
    }
}

// ---------------- gather updated codes + commitment loss ----------------
// z_q_st = z_e + sg(z_q - z_e) == z_q numerically.
__global__ __launch_bounds__(256) void vq_gather(
    const float* __restrict__ z, const int* __restrict__ idx,
    const float* __restrict__ newemb, float* __restrict__ out_zq,
    float* __restrict__ loss)
{
    __shared__ float red[256];
    const size_t t = (size_t)blockIdx.x * 256 + threadIdx.x;  // over B*D*L
    const int l = (int)(t & (LL - 1));
    const int d = (int)((t >> 12) & 63);
    const int b = (int)(t >> 18);
    const int n = b * LL + l;
    const float q  = newemb[(size_t)idx[n] * DD + d];
    const float zv = z[t];
    out_zq[t] = q;
    const float diff = q - zv;
    red[threadIdx.x] = diff * diff;
    __syncthreads();
    for (int s = 128; s > 0; s >>= 1) {
        if (threadIdx.x < s) red[threadIdx.x] += red[threadIdx.x + s];
        __syncthreads();
    }
    if (threadIdx.x == 0) atomicAdd(loss, red[0]);
}

// ---------------- scalars: vq_loss, perplexity ----------------
__global__ __launch_bounds__(512) void vq_finalize(
    const float* __restrict__ counts, const float* __restrict__ loss,
    float* __restrict__ out)
{
    __shared__ float red[512];
    const int k = threadIdx.x;
    const float c = counts[k];
    red[k] = c;
    __syncthreads();
    for (int s = 256; s > 0; s >>= 1) {
        if (k < s) red[k] += red[k + s];
        __syncthreads();
    }
    const float total = red[0];
    __syncthreads();
    const float p = c / total;
    red[k] = p * logf(p + 1e-10f);
    __syncthreads();
    for (int s = 256; s > 0; s >>= 1) {
        if (k < s) red[k] += red[k + s];
        __syncthreads();
    }
    if (k == 0) {
        out[OUT_LOSS] = BETAF * (loss[0] / (float)(BB * DD * LL));
        out[OUT_PERP] = expf(-red[0]);
    }
}

// ---------------- indices -> float tail of d_out ----------------
__global__ __launch_bounds__(256) void vq_idx2f(const int* __restrict__ idx,
                                                float* __restrict__ out)
{
    const int n = blockIdx.x * 256 + threadIdx.x;
    if (n < NR) out[OUT_IDX + n] = (float)idx[n];
}

extern "C" void kernel_launch(void* const* d_in, const int* in_sizes, int n_in,
                              void* d_out, int out_size, void* d_ws, size_t ws_size,
                              hipStream_t stream) {
    const float* z_e          = (const float*)d_in[0];   // [32, 64, 4096]
    const float* embedding    = (const float*)d_in[1];   // [512, 64]
    const float* cluster_size = (const float*)d_in[2];   // [512]
    const float* emb_avg      = (const float*)d_in[3];   // [512, 64]
    float* out = (float*)d_out;
    float* ws  = (float*)d_ws;

    float* e_norm = ws + WS_ENORM;
    float* counts = ws + WS_COUNTS;
    float* esum   = ws + WS_ESUM;
    float* newemb = ws + WS_NEWEMB;
    float* loss   = ws + WS_LOSS;
    int*   idxbuf = (int*)(ws + WS_IDX);

    // 1. zero accumulators (counts, esum, loss)
    vq_zero<<<(33281 + 255) / 256, 256, 0, stream>>>(ws);
    // 2. ||e||^2
    vq_enorm<<<2, 256, 0, stream>>>(embedding, e_norm);
    // 3. WMMA distances + argmin + counts    (8192 row-tiles, 8 waves/block)
    vq_argmin_wmma<<<NR / 16 / 8, 256, 0, stream>>>(z_e, embedding, e_norm,
                                                    idxbuf, counts);
    // 4. embed_sum scatter
    vq_scatter<<<(NR * DD) / 256, 256, 0, stream>>>(z_e, idxbuf, esum);
    // 5. EMA update -> new embedding
    vq_ema<<<1, 512, 0, stream>>>(cluster_size, emb_avg, counts, esum, newemb);
    // 6. gather + commitment loss
    vq_gather<<<(BB * DD * LL) / 256, 256, 0, stream>>>(z_e, idxbuf, newemb,
                                                        out + OUT_ZQ, loss);
    // 7. scalars
    vq_finalize<<<1, 512, 0, stream>>>(counts, loss, out);
    // 8. indices as float
    vq_idx2f<<<(NR + 255) / 256, 256, 0, stream>>>(idxbuf, out);
}